// HetGTAN_mean_76682346102825
// MI455X (gfx1250) — compile-verified
//
#include <hip/hip_runtime.h>

// ---------------- problem constants (match reference) ----------------
#define NA   50000
#define NP   100000
#define HIDD 64
#define OUTD 16
#define NEDGE 2000000
#define NHOP 3

typedef float v2f __attribute__((ext_vector_type(2)));
typedef float v8f __attribute__((ext_vector_type(8)));

// =====================================================================
// WMMA fp32 GEMM:  C[M,N] = act(A[M,K] @ W[K,N] + bias)
// act: 0 = identity, 1 = relu.  M%16==0, N%16==0, K%4==0.
// One wave32 computes one 16x16 output tile via V_WMMA_F32_16X16X4_F32.
// =====================================================================
__global__ void hetgtan_gemm_wmma_f32(const float* __restrict__ A,
                                      const float* __restrict__ W,
                                      const float* __restrict__ bias,
                                      float* __restrict__ C,
                                      int M, int K, int N, int act) {
    const int wavesPerBlock = blockDim.x >> 5;
    const int waveId = blockIdx.x * wavesPerBlock + (threadIdx.x >> 5);
    const int nTilesN = N >> 4;
    const int totalTiles = (M >> 4) * nTilesN;
    if (waveId >= totalTiles) return;          // wave-uniform: EXEC all-ones below

    const int mTile = waveId / nTilesN;
    const int nTile = waveId % nTilesN;
    const int lane  = threadIdx.x & 31;
    const int half  = lane >> 4;               // 0: K pair {0,1}, 1: K pair {2,3}
    const int l15   = lane & 15;

    const float* __restrict__ Arow = A + (long)(mTile * 16 + l15) * K;
    const float* __restrict__ Wcol = W + (nTile * 16 + l15);

    v8f acc = {};
    for (int k = 0; k < K; k += 4) {
        const int ka = k + half * 2;
        v2f a; a.x = Arow[ka];                 a.y = Arow[ka + 1];
        v2f b; b.x = Wcol[(long)ka * N];       b.y = Wcol[(long)(ka + 1) * N];
        // D = A(16x4) * B(4x16) + C  -> v_wmma_f32_16x16x4_f32
        acc = __builtin_amdgcn_wmma_f32_16x16x4_f32(
                  false, a, false, b, (short)0, acc, false, false);
    }

    const float bv = bias ? bias[nTile * 16 + l15] : 0.0f;
    #pragma unroll
    for (int r = 0; r < 8; ++r) {
        const int row = mTile * 16 + r + half * 8;
        float v = acc[r] + bv;
        if (act) v = v > 0.0f ? v : 0.0f;
        C[(long)row * N + nTile * 16 + l15] = v;
    }
}

// =====================================================================
// Per-source-node init for one relation (one wave32 per node):
//   x1[n]  = dot(x[n], a1)
//   w2     = exp(lrelu(x1[n] + dot(x[n], a2)))
//   div[n] = w2            (accumulator init: self term)
//   num[n] = w2 * x[n]     (accumulator init: self term)
// =====================================================================
__global__ void hetgtan_node_init_src(const float* __restrict__ x,
                                      const float* __restrict__ a1,
                                      const float* __restrict__ a2,
                                      float* __restrict__ x1,
                                      float* __restrict__ divv,
                                      float* __restrict__ num,
                                      int n) {
    const int wid = blockIdx.x * (blockDim.x >> 5) + (threadIdx.x >> 5);
    if (wid >= n) return;
    const int lane = threadIdx.x & 31;

    const float2 xv  = *(const float2*)(x  + (long)wid * HIDD + 2 * lane);
    const float2 a1v = *(const float2*)(a1 + 2 * lane);
    const float2 a2v = *(const float2*)(a2 + 2 * lane);

    float d1 = xv.x * a1v.x + xv.y * a1v.y;
    float d2 = xv.x * a2v.x + xv.y * a2v.y;
    #pragma unroll
    for (int o = 16; o > 0; o >>= 1) {
        d1 += __shfl_xor(d1, o, 32);
        d2 += __shfl_xor(d2, o, 32);
    }
    const float s  = d1 + d2;
    const float lr = s >= 0.0f ? s : 0.2f * s;
    const float w2 = __expf(lr);

    if (lane == 0) { x1[wid] = d1; divv[wid] = w2; }
    float2 nv; nv.x = w2 * xv.x; nv.y = w2 * xv.y;
    *(float2*)(num + (long)wid * HIDD + 2 * lane) = nv;
}

// h1[n] = dot(h[n], a2)   (one wave32 per node)
__global__ void hetgtan_node_h1(const float* __restrict__ h,
                                const float* __restrict__ a2,
                                float* __restrict__ h1,
                                int n) {
    const int wid = blockIdx.x * (blockDim.x >> 5) + (threadIdx.x >> 5);
    if (wid >= n) return;
    const int lane = threadIdx.x & 31;

    const float2 hv  = *(const float2*)(h  + (long)wid * HIDD + 2 * lane);
    const float2 a2v = *(const float2*)(a2 + 2 * lane);
    float d = hv.x * a2v.x + hv.y * a2v.y;
    #pragma unroll
    for (int o = 16; o > 0; o >>= 1) d += __shfl_xor(d, o, 32);
    if (lane == 0) h1[wid] = d;
}

// =====================================================================
// Edge scatter (one wave32 per edge, grid-stride):
//   w = exp(lrelu(x1[s] + h1[t]))
//   div[s]    += w           (lane 0)
//   num[s][:] += w * h[t][:] (float2 per lane, L2-resident fp32 atomics)
// =====================================================================
__global__ void hetgtan_edge_agg(const int* __restrict__ es,
                                 const int* __restrict__ et,
                                 const float* __restrict__ x1,
                                 const float* __restrict__ h1,
                                 const float* __restrict__ h,
                                 float* __restrict__ num,
                                 float* __restrict__ divv,
                                 int nEdges) {
    const int wavesPerBlock = blockDim.x >> 5;
    const int wid  = blockIdx.x * wavesPerBlock + (threadIdx.x >> 5);
    const int nW   = gridDim.x * wavesPerBlock;
    const int lane = threadIdx.x & 31;

    for (int e = wid; e < nEdges; e += nW) {
        const int s = es[e];
        const int t = et[e];
        const float sc = x1[s] + h1[t];          // uniform per wave -> broadcast
        const float lr = sc >= 0.0f ? sc : 0.2f * sc;
        const float w  = __expf(lr);

        const float2 hv = *(const float2*)(h + (long)t * HIDD + 2 * lane);
        float* np = num + (long)s * HIDD + 2 * lane;
        atomicAdd(np,     w * hv.x);
        atomicAdd(np + 1, w * hv.y);
        if (lane == 0) atomicAdd(divv + s, w);
    }
}

// h[i] = elu(num[i] / div[node])
__global__ void hetgtan_finalize_a(const float* __restrict__ num,
                                   const float* __restrict__ divv,
                                   float* __restrict__ h, int n) {
    const int i = blockIdx.x * blockDim.x + threadIdx.x;
    if (i >= n * HIDD) return;
    const float v = num[i] / divv[i >> 6];
    h[i] = v > 0.0f ? v : expm1f(v);
}

// h[i] = elu(0.5 * (num1[i]/div1[node] + num2[i]/div2[node]))
__global__ void hetgtan_finalize_p(const float* __restrict__ n1,
                                   const float* __restrict__ d1,
                                   const float* __restrict__ n2,
                                   const float* __restrict__ d2,
                                   float* __restrict__ h, int n) {
    const int i = blockIdx.x * blockDim.x + threadIdx.x;
    if (i >= n * HIDD) return;
    const int node = i >> 6;
    const float v = 0.5f * (n1[i] / d1[node] + n2[i] / d2[node]);
    h[i] = v > 0.0f ? v : expm1f(v);
}

// =====================================================================
extern "C" void kernel_launch(void* const* d_in, const int* in_sizes, int n_in,
                              void* d_out, int out_size, void* d_ws, size_t ws_size,
                              hipStream_t stream) {
    (void)in_sizes; (void)n_in; (void)out_size; (void)ws_size;

    const float* x_a     = (const float*)d_in[0];
    const float* x_p     = (const float*)d_in[1];
    const int*   ap_s    = (const int*)d_in[2];
    const int*   ap_t    = (const int*)d_in[3];
    const int*   pa_s    = (const int*)d_in[4];
    const int*   pa_t    = (const int*)d_in[5];
    const int*   pp_s    = (const int*)d_in[6];
    const int*   pp_t    = (const int*)d_in[7];
    const float* fc1_a_w = (const float*)d_in[8];
    const float* fc1_a_b = (const float*)d_in[9];
    const float* fc1_p_w = (const float*)d_in[10];
    const float* fc1_p_b = (const float*)d_in[11];
    const float* fc2_w   = (const float*)d_in[12];
    const float* fc2_b   = (const float*)d_in[13];
    const float* a1_ap   = (const float*)d_in[14];
    const float* a2_ap   = (const float*)d_in[15];
    const float* a1_pa   = (const float*)d_in[16];
    const float* a2_pa   = (const float*)d_in[17];
    const float* a1_pp   = (const float*)d_in[18];
    const float* a2_pp   = (const float*)d_in[19];
    float* out = (float*)d_out;

    // ---------------- workspace layout (floats) ----------------
    float* ws = (float*)d_ws;
    size_t off = 0;
    float* xa    = ws + off; off += (size_t)NA * HIDD;
    float* xp    = ws + off; off += (size_t)NP * HIDD;
    float* h_a   = ws + off; off += (size_t)NA * HIDD;
    float* h_p   = ws + off; off += (size_t)NP * HIDD;
    float* num_a = ws + off; off += (size_t)NA * HIDD;
    float* num_p1= ws + off; off += (size_t)NP * HIDD;
    float* num_p2= ws + off; off += (size_t)NP * HIDD;
    float* div_a = ws + off; off += NA;
    float* div_p1= ws + off; off += NP;
    float* div_p2= ws + off; off += NP;
    float* x1_a  = ws + off; off += NA;
    float* x1_p  = ws + off; off += NP;
    float* h1_a  = ws + off; off += NA;
    float* h1_p  = ws + off; off += NP;

    const int BLK = 256;                 // 8 wave32 per block
    const int WPB = BLK / 32;

    // -------- input projections (WMMA) --------
    {
        int tiles = (NA / 16) * (HIDD / 16);
        hetgtan_gemm_wmma_f32<<<(tiles + WPB - 1) / WPB, BLK, 0, stream>>>(
            x_a, fc1_a_w, fc1_a_b, xa, NA, 512, HIDD, 1);
        tiles = (NP / 16) * (HIDD / 16);
        hetgtan_gemm_wmma_f32<<<(tiles + WPB - 1) / WPB, BLK, 0, stream>>>(
            x_p, fc1_p_w, fc1_p_b, xp, NP, 256, HIDD, 1);
    }
    // h = x (copy)
    hipMemcpyAsync(h_a, xa, (size_t)NA * HIDD * sizeof(float),
                   hipMemcpyDeviceToDevice, stream);
    hipMemcpyAsync(h_p, xp, (size_t)NP * HIDD * sizeof(float),
                   hipMemcpyDeviceToDevice, stream);

    const int nodeBlocksA = (NA + WPB - 1) / WPB;
    const int nodeBlocksP = (NP + WPB - 1) / WPB;
    const int edgeBlocks  = 6144;        // grid-stride, ~49K waves
    const int finBlocksA  = (NA * HIDD + BLK - 1) / BLK;
    const int finBlocksP  = (NP * HIDD + BLK - 1) / BLK;

    for (int i = 0; i < NHOP; ++i) {
        const float* a1ap = a1_ap + i * HIDD;  const float* a2ap = a2_ap + i * HIDD;
        const float* a1pa = a1_pa + i * HIDD;  const float* a2pa = a2_pa + i * HIDD;
        const float* a1pp = a1_pp + i * HIDD;  const float* a2pp = a2_pp + i * HIDD;

        // relation a->p : x = xa (source a), h = h_p (target p)
        hetgtan_node_init_src<<<nodeBlocksA, BLK, 0, stream>>>(
            xa, a1ap, a2ap, x1_a, div_a, num_a, NA);
        hetgtan_node_h1<<<nodeBlocksP, BLK, 0, stream>>>(h_p, a2ap, h1_p, NP);
        hetgtan_edge_agg<<<edgeBlocks, BLK, 0, stream>>>(
            ap_s, ap_t, x1_a, h1_p, h_p, num_a, div_a, NEDGE);

        // relation p->a : x = xp (source p), h = h_a (target a)
        hetgtan_node_init_src<<<nodeBlocksP, BLK, 0, stream>>>(
            xp, a1pa, a2pa, x1_p, div_p1, num_p1, NP);
        hetgtan_node_h1<<<nodeBlocksA, BLK, 0, stream>>>(h_a, a2pa, h1_a, NA);
        hetgtan_edge_agg<<<edgeBlocks, BLK, 0, stream>>>(
            pa_s, pa_t, x1_p, h1_a, h_a, num_p1, div_p1, NEDGE);

        // h_a no longer read this hop -> finalize: h_a = elu(agg_a)
        hetgtan_finalize_a<<<finBlocksA, BLK, 0, stream>>>(num_a, div_a, h_a, NA);

        // relation p->p : x = xp (source p), h = h_p (target p)
        hetgtan_node_init_src<<<nodeBlocksP, BLK, 0, stream>>>(
            xp, a1pp, a2pp, x1_p, div_p2, num_p2, NP);
        hetgtan_node_h1<<<nodeBlocksP, BLK, 0, stream>>>(h_p, a2pp, h1_p, NP);
        hetgtan_edge_agg<<<edgeBlocks, BLK, 0, stream>>>(
            pp_s, pp_t, x1_p, h1_p, h_p, num_p2, div_p2, NEDGE);

        // h_p = elu(0.5*(agg_p1 + agg_p2))
        hetgtan_finalize_p<<<finBlocksP, BLK, 0, stream>>>(
            num_p1, div_p1, num_p2, div_p2, h_p, NP);
    }

    // -------- output projection (WMMA) straight into d_out --------
    {
        int tiles = (NA / 16) * (OUTD / 16);
        hetgtan_gemm_wmma_f32<<<(tiles + WPB - 1) / WPB, BLK, 0, stream>>>(
            h_a, fc2_w, fc2_b, out, NA, HIDD, OUTD, 0);
    }
}